// SparseConv2d_13030930776769
// MI455X (gfx1250) — compile-verified
//
#include <hip/hip_runtime.h>

#define Bn 4
#define Hn 1024
#define Wn 1024
#define Nn 1000000
#define CIN 64
#define COUT 64
#define TILE_M 128
#define NTAP 9
#define THREADS 256

typedef __attribute__((ext_vector_type(2))) float v2f;
typedef __attribute__((ext_vector_type(8))) float v8f;

// ---- kernel 1: fill idx_map with -1 (int4 vectorized) ----
__global__ __launch_bounds__(THREADS) void init_map_kernel(int4* __restrict__ map) {
  int i = blockIdx.x * blockDim.x + threadIdx.x;
  map[i] = make_int4(-1, -1, -1, -1);
}

// ---- kernel 2: scatter point ids into idx_map ----
__global__ __launch_bounds__(THREADS) void scatter_kernel(const int* __restrict__ idx,
                                                          int* __restrict__ map, int n) {
  int i = blockIdx.x * blockDim.x + threadIdx.x;
  if (i < n) {
    int b = idx[i * 3 + 0];
    int y = idx[i * 3 + 1];
    int x = idx[i * 3 + 2];
    map[(b * Hn + y) * Wn + x] = i;
  }
}

// ---- kernel 3: permute weights into WMMA B-fragment order ----
// fragment layout: wf[(((t*4 + nt)*16 + ks)*64) + lane*2 + j]
//   lane<16 : b[j] = W[t][k=ks*4+j    ][n=nt*16+lane]
//   lane>=16: b[j] = W[t][k=ks*4+2+j  ][n=nt*16+lane-16]
__global__ __launch_bounds__(THREADS) void swizzle_w_kernel(const float* __restrict__ w,
                                                            float* __restrict__ wf) {
  int e = blockIdx.x * blockDim.x + threadIdx.x;   // 0 .. 9*64*64-1
  int j    = e & 1;
  int lane = (e >> 1) & 31;
  int ks   = (e >> 6) & 15;
  int nt   = (e >> 10) & 3;
  int t    = e >> 12;
  int k = ks * 4 + ((lane >> 4) << 1) + j;
  int n = nt * 16 + (lane & 15);
  wf[e] = w[(t * CIN + k) * COUT + n];
}

// ---- kernel 4: gather-GEMM sparse conv using V_WMMA_F32_16X16X4_F32 ----
__global__ __launch_bounds__(THREADS) void conv_kernel(
    const float* __restrict__ feat,
    const float* __restrict__ bias,
    const int*   __restrict__ indices,
    const int*   __restrict__ idx_map,
    const float* __restrict__ wfrag,
    float*       __restrict__ out) {
  __shared__ float sA[TILE_M * 68];      // gathered feature tile, padded rows (34.8 KB)
  __shared__ float sW[CIN * 16 * 4];     // one tap's fragment-ordered weights (16 KB)
  __shared__ int   sN[TILE_M * NTAP];    // neighbor ids (4.6 KB)

  const int tid  = threadIdx.x;
  const int wv   = tid >> 5;       // wave id 0..7 == M-tile id
  const int lane = tid & 31;
  const int base = blockIdx.x * TILE_M;

  // ---- neighbor index table for this block (idx_map is L2-resident) ----
  for (int i = tid; i < TILE_M * NTAP; i += THREADS) {
    int m = i / NTAP;
    int t = i - m * NTAP;
    int p = base + m;
    int nv = -1;
    if (p < Nn) {
      int b = indices[p * 3 + 0];
      int y = indices[p * 3 + 1];
      int x = indices[p * 3 + 2];
      int ny = y + (t / 3) - 1;
      int nx = x + (t % 3) - 1;
      if (ny >= 0 && ny < Hn && nx >= 0 && nx < Wn)
        nv = idx_map[(b * Hn + ny) * Wn + nx];
    }
    sN[i] = nv;
  }

  // ---- accumulators seeded with bias (C/D layout: vgpr r -> M=r(+8), N=lane&15) ----
  v8f acc[4];
  for (int nt = 0; nt < 4; nt++) {
    float bv = bias[nt * 16 + (lane & 15)];
#pragma unroll
    for (int r = 0; r < 8; r++) acc[nt][r] = bv;
  }

  // A-fragment base: lane L reads row (wv*16 + L&15), K offset 2*(L>>4), stride 4 per ks
  const int arow = wv * 16 + (lane & 15);
  const float* aB = &sA[arow * 68 + ((lane >> 4) << 1)];
  const float* wB = &sW[lane * 2];

  __syncthreads();

  for (int t = 0; t < NTAP; t++) {
    // gather 128 rows x 64ch of features (float4, zero-fill missing neighbors)
    for (int i = tid; i < TILE_M * 16; i += THREADS) {
      int m = i >> 4;
      int c = i & 15;
      int nv = sN[m * NTAP + t];
      float4 v = make_float4(0.f, 0.f, 0.f, 0.f);
      if (nv >= 0) v = ((const float4*)(feat + (size_t)nv * CIN))[c];
      *(float4*)&sA[m * 68 + c * 4] = v;
    }
    // copy this tap's pre-swizzled weight fragments (coalesced, L2-hot)
    const float4* ws = (const float4*)(wfrag + t * (CIN * 16 * 4));
    for (int i = tid; i < 1024; i += THREADS) ((float4*)sW)[i] = ws[i];
    __syncthreads();

    // 16 K-steps x 4 N-tiles of V_WMMA_F32_16X16X4_F32 (uniform flow, EXEC all-1s)
#pragma unroll
    for (int ks = 0; ks < 16; ks++) {
      v2f a = *(const v2f*)(aB + ks * 4);
#pragma unroll
      for (int nt = 0; nt < 4; nt++) {
        v2f b = *(const v2f*)(wB + (nt * 16 + ks) * 64);
        acc[nt] = __builtin_amdgcn_wmma_f32_16x16x4_f32(
            false, a, false, b, (short)0, acc[nt], false, false);
      }
    }
    __syncthreads();
  }

  // ---- store D: vgpr r -> row base+wv*16+r+(lane>=16?8:0), col nt*16+(lane&15) ----
  for (int nt = 0; nt < 4; nt++) {
    int n  = nt * 16 + (lane & 15);
    int mh = (lane >> 4) << 3;
#pragma unroll
    for (int r = 0; r < 8; r++) {
      int p = base + wv * 16 + mh + r;
      if (p < Nn)
        __builtin_nontemporal_store(acc[nt][r], &out[(size_t)p * COUT + n]);
    }
  }
}

extern "C" void kernel_launch(void* const* d_in, const int* in_sizes, int n_in,
                              void* d_out, int out_size, void* d_ws, size_t ws_size,
                              hipStream_t stream) {
  const float* feat   = (const float*)d_in[0];
  const float* weight = (const float*)d_in[1];
  const float* bias   = (const float*)d_in[2];
  const int*   idx    = (const int*)d_in[3];
  float* out = (float*)d_out;

  int*   map = (int*)d_ws;                                          // 16.78 MB
  float* wf  = (float*)((char*)d_ws + (size_t)Bn * Hn * Wn * 4);    // +147 KB

  init_map_kernel<<<(Bn * Hn * Wn / 4) / THREADS, THREADS, 0, stream>>>((int4*)map);
  scatter_kernel<<<(Nn + THREADS - 1) / THREADS, THREADS, 0, stream>>>(idx, map, Nn);
  swizzle_w_kernel<<<(NTAP * CIN * COUT) / THREADS, THREADS, 0, stream>>>(weight, wf);
  conv_kernel<<<(Nn + TILE_M - 1) / TILE_M, THREADS, 0, stream>>>(feat, bias, idx, map, wf, out);
}